// SCOUT_73418170958485
// MI455X (gfx1250) — compile-verified
//
#include <hip/hip_runtime.h>
#include <hip/hip_bf16.h>

// ---------------------------------------------------------------------------
// GAT layer for gfx1250 (MI455X).  fp32 end-to-end; GEMMs on
// V_WMMA_F32_16X16X4_F32 (wave32 WMMA).  Edge phase is wave-per-edge with
// float atomics (agg/z fit in the 192MB L2, so atomics/gathers stay on-chip).
// ---------------------------------------------------------------------------

typedef float v2f __attribute__((ext_vector_type(2)));
typedef float v8f __attribute__((ext_vector_type(8)));

#define D_FEAT 128
#define NEG_SLOPE 0.01f

// Monotone order-preserving float->uint key (for segment-max via atomicMax).
__device__ __forceinline__ unsigned fkey(float f) {
    unsigned u = __float_as_uint(f);
    return (u & 0x80000000u) ? ~u : (u | 0x80000000u);
}
__device__ __forceinline__ float fdec(unsigned k) {
    return __uint_as_float((k & 0x80000000u) ? (k ^ 0x80000000u) : ~k);
}

__device__ __forceinline__ float wave32_sum(float v) {
    #pragma unroll
    for (int off = 16; off > 0; off >>= 1) v += __shfl_xor(v, off, 32);
    return v;
}

// ---------------------------------------------------------------------------
// K0: zero scratch (agg | denom | m_key are laid out contiguously)
// ---------------------------------------------------------------------------
__global__ __launch_bounds__(256) void k_zero(float* __restrict__ p, size_t n) {
    size_t i = (size_t)blockIdx.x * blockDim.x + threadIdx.x;
    if (i < n) p[i] = 0.0f;
}

// ---------------------------------------------------------------------------
// K1: dual GEMM  hs = h @ Wself^T,  z = h @ Wfunc^T   via WMMA f32 16x16x4.
// One wave (32 lanes) per 16-row tile; blockIdx.y selects which weight/output.
// A frag: float2 = h[row][kk + 2*half .. +1]          (row = rt*16 + lane&15)
// B frag: float2 = W[col][kk + 2*half .. +1]  (B=W^T)  (col = ct*16 + lane&15)
// D frag: VGPR j -> row (j + 8*half), col = lane&15.
// ---------------------------------------------------------------------------
__global__ __launch_bounds__(32) void k_gemm_wmma(
    const float* __restrict__ h, const float* __restrict__ Wself,
    const float* __restrict__ Wfunc, float* __restrict__ hs,
    float* __restrict__ z, int N)
{
    const int rt   = blockIdx.x;
    const float* W = blockIdx.y ? Wfunc : Wself;
    float* out     = blockIdx.y ? z : hs;
    const int lane = threadIdx.x;
    const int half = lane >> 4;       // 0: lanes 0-15, 1: lanes 16-31
    const int l16  = lane & 15;

    int row = rt * 16 + l16;
    if (row >= N) row = N - 1;        // duplicate last row; stores are guarded
    const float* arow = h + (size_t)row * D_FEAT + 2 * half;

    v8f acc[8] = {};                  // 8 column tiles of 16  (D = 128)

    for (int kk = 0; kk < D_FEAT; kk += 4) {
        v2f a = *(const v2f*)(arow + kk);
        #pragma unroll
        for (int ct = 0; ct < 8; ++ct) {
            const float* bp = W + (size_t)(ct * 16 + l16) * D_FEAT + kk + 2 * half;
            v2f b = *(const v2f*)bp;
            acc[ct] = __builtin_amdgcn_wmma_f32_16x16x4_f32(
                false, a, false, b, (short)0, acc[ct], false, false);
        }
    }

    #pragma unroll
    for (int ct = 0; ct < 8; ++ct) {
        #pragma unroll
        for (int j = 0; j < 8; ++j) {
            int r = rt * 16 + j + 8 * half;
            if (r < N) out[(size_t)r * D_FEAT + ct * 16 + l16] = acc[ct][j];
        }
    }
}

// ---------------------------------------------------------------------------
// K2: per-node attention scores  s_src = z.wa[0:128], s_dst = z.wa[128:256]
// One wave per node; float4 per lane; shfl reduction.
// ---------------------------------------------------------------------------
__global__ __launch_bounds__(256) void k_node_scores(
    const float* __restrict__ z, const float* __restrict__ Watt,
    float* __restrict__ s_src, float* __restrict__ s_dst, int N)
{
    int wv   = (int)(((size_t)blockIdx.x * blockDim.x + threadIdx.x) >> 5);
    int lane = threadIdx.x & 31;
    if (wv >= N) return;
    float4 zv = *(const float4*)(z + (size_t)wv * D_FEAT + lane * 4);
    float4 wa = *(const float4*)(Watt + lane * 4);
    float4 wb = *(const float4*)(Watt + D_FEAT + lane * 4);
    float sa = zv.x * wa.x + zv.y * wa.y + zv.z * wa.z + zv.w * wa.w;
    float sb = zv.x * wb.x + zv.y * wb.y + zv.z * wb.z + zv.w * wb.w;
    sa = wave32_sum(sa);
    sb = wave32_sum(sb);
    if (lane == 0) { s_src[wv] = sa; s_dst[wv] = sb; }
}

// ---------------------------------------------------------------------------
// K3: per-edge logit = leaky_relu(s_src[src] + s_dst[dst] + edge_w.wa[256:])
//     + segment-max over dst via uint-key atomicMax.  One wave per edge.
// ---------------------------------------------------------------------------
__global__ __launch_bounds__(256) void k_edge_logits(
    const float* __restrict__ edge_w, const float* __restrict__ Watt,
    const float* __restrict__ s_src, const float* __restrict__ s_dst,
    const int* __restrict__ src, const int* __restrict__ dst,
    float* __restrict__ logits, unsigned* __restrict__ m_key, int E)
{
    int e    = (int)(((size_t)blockIdx.x * blockDim.x + threadIdx.x) >> 5);
    int lane = threadIdx.x & 31;
    if (e >= E) return;
    float4 ev = *(const float4*)(edge_w + (size_t)e * D_FEAT + lane * 4);
    float4 we = *(const float4*)(Watt + 2 * D_FEAT + lane * 4);
    float p = ev.x * we.x + ev.y * we.y + ev.z * we.z + ev.w * we.w;
    p = wave32_sum(p);
    if (lane == 0) {
        float lg = s_src[src[e]] + s_dst[dst[e]] + p;
        lg = (lg > 0.0f) ? lg : NEG_SLOPE * lg;
        logits[e] = lg;
        atomicMax(m_key + dst[e], fkey(lg));
    }
}

// ---------------------------------------------------------------------------
// K4: denom[dst] += exp(logit - m[dst]).  One thread per edge.
// ---------------------------------------------------------------------------
__global__ __launch_bounds__(256) void k_edge_denom(
    const float* __restrict__ logits, const int* __restrict__ dst,
    const unsigned* __restrict__ m_key, float* __restrict__ denom, int E)
{
    int e = (int)((size_t)blockIdx.x * blockDim.x + threadIdx.x);
    if (e >= E) return;
    int d = dst[e];
    atomicAdd(denom + d, expf(logits[e] - fdec(m_key[d])));
}

// ---------------------------------------------------------------------------
// K5: agg[dst] += a * z[src],  a = exp(logit - m)/max(denom,1e-9).
// One wave per edge; 4 floats per lane; atomics land in L2 (agg = 25.6MB).
// ---------------------------------------------------------------------------
__global__ __launch_bounds__(256) void k_edge_scatter(
    const float* __restrict__ z, const float* __restrict__ logits,
    const int* __restrict__ src, const int* __restrict__ dst,
    const unsigned* __restrict__ m_key, const float* __restrict__ denom,
    float* __restrict__ agg, int E)
{
    int e    = (int)(((size_t)blockIdx.x * blockDim.x + threadIdx.x) >> 5);
    int lane = threadIdx.x & 31;
    if (e >= E) return;
    int s = src[e], d = dst[e];
    float a = expf(logits[e] - fdec(m_key[d])) / fmaxf(denom[d], 1e-9f);
    float4 zv = *(const float4*)(z + (size_t)s * D_FEAT + lane * 4);
    float* ap = agg + (size_t)d * D_FEAT + lane * 4;
    atomicAdd(ap + 0, a * zv.x);
    atomicAdd(ap + 1, a * zv.y);
    atomicAdd(ap + 2, a * zv.z);
    atomicAdd(ap + 3, a * zv.w);
}

// ---------------------------------------------------------------------------
// K6: out = h_in + relu(deg>0 ? hs+agg : h_in).   denom>0 <=> deg>0
// (every edge contributes exp(logit-m) and the per-group max contributes 1.0).
// ---------------------------------------------------------------------------
__global__ __launch_bounds__(256) void k_finalize(
    const float* __restrict__ h, const float* __restrict__ hs,
    const float* __restrict__ agg, const float* __restrict__ denom,
    float* __restrict__ out, size_t ND)
{
    size_t i = (size_t)blockIdx.x * blockDim.x + threadIdx.x;
    if (i >= ND) return;
    size_t node = i >> 7;                    // D_FEAT = 128
    float hin = h[i];
    float v = (denom[node] > 0.0f) ? (hs[i] + agg[i]) : hin;
    v = fmaxf(v, 0.0f);
    out[i] = hin + v;
}

// ---------------------------------------------------------------------------
extern "C" void kernel_launch(void* const* d_in, const int* in_sizes, int n_in,
                              void* d_out, int out_size, void* d_ws, size_t ws_size,
                              hipStream_t stream) {
    const float* h      = (const float*)d_in[0];
    const float* edge_w = (const float*)d_in[1];
    const float* Wself  = (const float*)d_in[2];
    const float* Wfunc  = (const float*)d_in[3];
    const float* Watt   = (const float*)d_in[4];
    const int*   src    = (const int*)d_in[5];
    const int*   dst    = (const int*)d_in[6];
    const int N = in_sizes[0] / D_FEAT;
    const int E = in_sizes[5];
    float* out = (float*)d_out;

    // Workspace layout (floats):  hs | z | s_src | s_dst | logits | agg | denom | m_key
    const size_t ND = (size_t)N * D_FEAT;
    float*    ws     = (float*)d_ws;
    float*    hs     = ws;
    float*    z      = hs + ND;
    float*    s_src  = z + ND;
    float*    s_dst  = s_src + N;
    float*    logits = s_dst + N;
    float*    agg    = logits + E;
    float*    denom  = agg + ND;
    unsigned* m_key  = (unsigned*)(denom + N);   // zero-bits == smallest key

    const size_t zero_n = ND + 2 * (size_t)N;    // agg + denom + m_key
    k_zero<<<(unsigned)((zero_n + 255) / 256), 256, 0, stream>>>(agg, zero_n);

    dim3 gemm_grid((N + 15) / 16, 2);
    k_gemm_wmma<<<gemm_grid, 32, 0, stream>>>(h, Wself, Wfunc, hs, z, N);

    size_t nthN = (size_t)N * 32;
    k_node_scores<<<(unsigned)((nthN + 255) / 256), 256, 0, stream>>>(z, Watt, s_src, s_dst, N);

    size_t nthE = (size_t)E * 32;
    k_edge_logits<<<(unsigned)((nthE + 255) / 256), 256, 0, stream>>>(
        edge_w, Watt, s_src, s_dst, src, dst, logits, m_key, E);

    k_edge_denom<<<(unsigned)(((size_t)E + 255) / 256), 256, 0, stream>>>(
        logits, dst, m_key, denom, E);

    k_edge_scatter<<<(unsigned)((nthE + 255) / 256), 256, 0, stream>>>(
        z, logits, src, dst, m_key, denom, agg, E);

    k_finalize<<<(unsigned)((ND + 255) / 256), 256, 0, stream>>>(h, hs, agg, denom, out, ND);
}